// DroneModel_19670950215788
// MI455X (gfx1250) — compile-verified
//
#include <hip/hip_runtime.h>

#define HORIZON   20
#define STATE_DIM 16
#define BATCH_N   524288

// Native clang vector type (works with __builtin_nontemporal_store)
typedef float v4f __attribute__((ext_vector_type(4)));

// Physics constants (match reference module constants, evaluated in f32)
constexpr float DT     = 0.05f;
constexpr float GRAV   = 9.81f;
constexpr float TAU_M  = 0.05f;
constexpr float MAX_T  = 5.0f;
constexpr float ARM    = 0.2f;
constexpr float KAPPA  = 0.02f;
constexpr float IX = 0.01f, IY = 0.01f, IZ = 0.02f;
constexpr float INV_IX = 1.0f / IX;
constexpr float INV_IY = 1.0f / IY;
constexpr float INV_IZ = 1.0f / IZ;
constexpr float DT_OVER_TAU = DT / TAU_M;   // == 1.0f, kept symbolic to mirror reference

__global__ __launch_bounds__(256) void drone_rollout_kernel(
    const float* __restrict__ state0,   // [BATCH, 16]
    const float* __restrict__ U,        // [HORIZON, 1, 4]
    float* __restrict__ out)            // [HORIZON, BATCH, 16]
{
    // ---- Stage sigmoid(U) into LDS once per block (80 values) ----
    __shared__ float sAct[HORIZON * 4];
    const int tid = threadIdx.x;
    if (tid < HORIZON * 4) {
        float u = U[tid];
        sAct[tid] = 1.0f / (1.0f + __expf(-u));
    }
    __syncthreads();

    const size_t b = (size_t)blockIdx.x * (size_t)blockDim.x + (size_t)tid;

    // ---- Load 16-float state into registers (4 x b128) ----
    const v4f* in4 = reinterpret_cast<const v4f*>(state0 + b * STATE_DIM);
    v4f s0 = in4[0];
    v4f s1 = in4[1];
    v4f s2 = in4[2];
    v4f s3 = in4[3];

    float px = s0.x, py = s0.y, pz = s0.z;       // pos
    float vx = s0.w, vy = s1.x, vz = s1.y;       // vel
    float phi = s1.z, th = s1.w, psi = s2.x;     // euler
    float wp = s2.y, wq = s2.z, wr = s2.w;       // body rates
    float t0 = s3.x, t1 = s3.y, t2 = s3.z, t3 = s3.w;  // rotor thrusts

    float* const outBase = out + b * STATE_DIM;

#pragma unroll
    for (int t = 0; t < HORIZON; ++t) {
        // actions for this step (uniform across batch, broadcast from LDS)
        const float a0 = sAct[4 * t + 0];
        const float a1 = sAct[4 * t + 1];
        const float a2 = sAct[4 * t + 2];
        const float a3 = sAct[4 * t + 3];

        // rotor first-order lag: thr_new = thr + DT/TAU_M * (a*MAX_T - thr)
        t0 = t0 + DT_OVER_TAU * (a0 * MAX_T - t0);
        t1 = t1 + DT_OVER_TAU * (a1 * MAX_T - t1);
        t2 = t2 + DT_OVER_TAU * (a2 * MAX_T - t2);
        t3 = t3 + DT_OVER_TAU * (a3 * MAX_T - t3);

        const float F    = t0 + t1 + t2 + t3;
        const float tau0 = ARM   * (t0 - t2);
        const float tau1 = ARM   * (t1 - t3);
        const float tau2 = KAPPA * (t0 - t1 + t2 - t3);

        // hardware trans ops: v_sin_f32 / v_cos_f32
        float sphi, cphi, sth, cth, spsi, cpsi;
        __sincosf(phi, &sphi, &cphi);
        __sincosf(th,  &sth,  &cth);
        __sincosf(psi, &spsi, &cpsi);

        // body z-axis in world frame
        const float zb0 = cphi * sth * cpsi + sphi * spsi;
        const float zb1 = cphi * sth * spsi - sphi * cpsi;
        const float zb2 = cphi * cth;

        // acceleration (MASS == 1)
        const float ax = F * zb0;
        const float ay = F * zb1;
        const float az = F * zb2 - GRAV;

        // euler-rate kinematics (v_rcp_f32 for 1/cos(theta))
        const float rcth  = __builtin_amdgcn_rcpf(cth);
        const float tanth = sth * rcth;
        const float ad0 = wp + sphi * tanth * wq + cphi * tanth * wr;
        const float ad1 = cphi * wq - sphi * wr;
        const float ad2 = (sphi * wq + cphi * wr) * rcth;

        // omega_dot = (tau - omega x (I*omega)) / I
        const float b0 = IX * wp, b1 = IY * wq, b2 = IZ * wr;
        const float c0 = wq * b2 - wr * b1;
        const float c1 = wr * b0 - wp * b2;
        const float c2 = wp * b1 - wq * b0;
        const float wd0 = (tau0 - c0) * INV_IX;
        const float wd1 = (tau1 - c1) * INV_IY;
        const float wd2 = (tau2 - c2) * INV_IZ;

        // Euler integrate (pos uses OLD vel: update order matters)
        px += DT * vx;  py += DT * vy;  pz += DT * vz;
        vx += DT * ax;  vy += DT * ay;  vz += DT * az;
        phi += DT * ad0; th += DT * ad1; psi += DT * ad2;
        wp += DT * wd0;  wq += DT * wd1;  wr += DT * wd2;

        // ---- Stream out state for this step: non-temporal b128 stores ----
        v4f* op = reinterpret_cast<v4f*>(outBase + (size_t)t * (size_t)BATCH_N * STATE_DIM);
        v4f o0; o0.x = px;  o0.y = py;  o0.z = pz;  o0.w = vx;
        v4f o1; o1.x = vy;  o1.y = vz;  o1.z = phi; o1.w = th;
        v4f o2; o2.x = psi; o2.y = wp;  o2.z = wq;  o2.w = wr;
        v4f o3; o3.x = t0;  o3.y = t1;  o3.z = t2;  o3.w = t3;
        __builtin_nontemporal_store(o0, op + 0);
        __builtin_nontemporal_store(o1, op + 1);
        __builtin_nontemporal_store(o2, op + 2);
        __builtin_nontemporal_store(o3, op + 3);
    }
}

extern "C" void kernel_launch(void* const* d_in, const int* in_sizes, int n_in,
                              void* d_out, int out_size, void* d_ws, size_t ws_size,
                              hipStream_t stream) {
    const float* current_state = (const float*)d_in[0];  // [BATCH, 16] f32
    const float* U             = (const float*)d_in[1];  // [HORIZON, 1, 4] f32
    float* out                 = (float*)d_out;          // [HORIZON, BATCH, 16] f32

    (void)in_sizes; (void)n_in; (void)out_size; (void)d_ws; (void)ws_size;

    const int block = 256;                         // 8 wave32s
    const int grid  = BATCH_N / block;             // 2048 blocks
    drone_rollout_kernel<<<grid, block, 0, stream>>>(current_state, U, out);
}